// norm_DotProductAttention_22892175687918
// MI455X (gfx1250) — compile-verified
//
#include <hip/hip_runtime.h>
#include <hip/hip_bf16.h>
#include <cmath>

#define NUM_NODES 50000
#define NUM_FEATS 64
#define NUM_EDGES 1600000
#define NUM_HEADS 2
#define QK_COLS 256  // NUM_HEADS * 2 * NUM_FEATS

typedef __attribute__((ext_vector_type(2))) float v2f;
typedef __attribute__((ext_vector_type(8))) float v8f;

// ---------------------------------------------------------------------------
// Sign-aware float atomic max via integer atomics (order-independent, exact).
// Works because positive floats order as ints, negative floats order reversed
// as unsigned ints. m is initialized to -inf (0xFF800000).
// ---------------------------------------------------------------------------
__device__ inline void atomicMaxFloat(float* addr, float val) {
  if (val >= 0.0f) {
    atomicMax((int*)addr, __float_as_int(val));
  } else {
    atomicMin((unsigned int*)addr, __float_as_uint(val));
  }
}

// ---------------------------------------------------------------------------
// qk = x @ W + b  using v_wmma_f32_16x16x4_f32 (fp32 WMMA: exact match to the
// fp32 reference). One wave per 16x16 output tile; K=64 in steps of 4.
//
// VGPR layouts per CDNA5 ISA 7.12.2:
//   A 16x4 f32 : lanes 0-15 M=0..15 hold K={kk,kk+1}; lanes 16-31 hold K={kk+2,kk+3}
//   B 4x16 f32 : VGPR0 rows K=kk (lanes0-15) / kk+2 (lanes16-31); VGPR1 = +1
//   C 16x16 f32: VGPR r -> M=r (lanes0-15), M=r+8 (lanes16-31), N = lane&15
// ---------------------------------------------------------------------------
__global__ __launch_bounds__(256) void gemm_qk_wmma(
    const float* __restrict__ x, const float* __restrict__ W,
    const float* __restrict__ bias, float* __restrict__ qk)
{
  const int wave = blockIdx.x * (blockDim.x >> 5) + (threadIdx.x >> 5);
  const int lane = threadIdx.x & 31;
  const int NT = QK_COLS / 16;            // 16 column tiles
  const int mt = wave / NT;
  const int nt = wave % NT;
  if (mt >= NUM_NODES / 16) return;       // 50000 divisible by 16 -> 3125 tiles
  const int m0 = mt * 16, n0 = nt * 16;
  const int half = lane >> 4;             // 0: lanes 0-15, 1: lanes 16-31
  const int lm   = lane & 15;

  v8f c = {};
  const float* arow = x + (size_t)(m0 + lm) * NUM_FEATS;
#pragma unroll
  for (int kk = 0; kk < NUM_FEATS; kk += 4) {
    v2f a;
    a.x = arow[kk + 2 * half + 0];
    a.y = arow[kk + 2 * half + 1];
    v2f b;
    b.x = W[(size_t)(kk + 2 * half + 0) * QK_COLS + n0 + lm];
    b.y = W[(size_t)(kk + 2 * half + 1) * QK_COLS + n0 + lm];
    // 8 args: (neg_a, A, neg_b, B, c_mod, C, reuse_a, reuse_b)
    c = __builtin_amdgcn_wmma_f32_16x16x4_f32(false, a, false, b,
                                              (short)0, c, false, false);
  }
  const float bv = bias[n0 + lm];
#pragma unroll
  for (int r = 0; r < 8; ++r) {
    const int rowm = m0 + r + 8 * half;
    qk[(size_t)rowm * QK_COLS + n0 + lm] = c[r] + bv;
  }
}

// ---------------------------------------------------------------------------
// Init per-(node,head) running max (-inf) and sum (0).
// ---------------------------------------------------------------------------
__global__ void init_ms(float* __restrict__ m, float* __restrict__ s) {
  int i = blockIdx.x * blockDim.x + threadIdx.x;
  if (i < NUM_NODES * NUM_HEADS) {
    m[i] = -INFINITY;
    s[i] = 0.0f;
  }
}

// ---------------------------------------------------------------------------
// Per-(edge,head) 64-wide dot product. 16 lanes per work item, float4 per
// lane (group loads 256 contiguous bytes -> L2-resident gather). Butterfly
// reduce within the 16-lane group, lane 0 writes score + atomic segment max.
// ---------------------------------------------------------------------------
__global__ __launch_bounds__(256) void edge_scores_kernel(
    const float* __restrict__ qk, const int* __restrict__ ei,
    float* __restrict__ scores, float* __restrict__ mbuf)
{
  const long long gid = (long long)blockIdx.x * blockDim.x + threadIdx.x;
  const long long group = gid >> 4;
  const int lane = (int)(gid & 15);
  if (group >= (long long)NUM_EDGES * NUM_HEADS) return;
  const int e = (int)(group >> 1);
  const int h = (int)(group & 1);
  const int r = ei[e];
  const int cc = ei[NUM_EDGES + e];

  const float4 qv = ((const float4*)(qk + (size_t)r  * QK_COLS + h * 128))[lane];
  const float4 kv = ((const float4*)(qk + (size_t)cc * QK_COLS + h * 128 + 64))[lane];
  float acc = qv.x * kv.x + qv.y * kv.y + qv.z * kv.z + qv.w * kv.w;
  acc += __shfl_xor(acc, 8, 32);   // stays inside 16-lane group (wave32)
  acc += __shfl_xor(acc, 4, 32);
  acc += __shfl_xor(acc, 2, 32);
  acc += __shfl_xor(acc, 1, 32);
  if (lane == 0) {
    scores[(size_t)e * NUM_HEADS + h] = acc;
    atomicMaxFloat(&mbuf[(size_t)r * NUM_HEADS + h], acc);
  }
}

// ---------------------------------------------------------------------------
// p = exp(score - m[row]); overwrite scores with p; segment-sum via atomics.
// ---------------------------------------------------------------------------
__global__ void exp_sum_kernel(const int* __restrict__ ei,
                               const float* __restrict__ mbuf,
                               float* __restrict__ scores,
                               float* __restrict__ sbuf)
{
  int i = blockIdx.x * blockDim.x + threadIdx.x;   // i = e*2 + h
  if (i >= NUM_EDGES * NUM_HEADS) return;
  int e = i >> 1, h = i & 1;
  int r = ei[e];
  float p = __expf(scores[i] - mbuf[r * NUM_HEADS + h]);
  scores[i] = p;
  atomicAdd(&sbuf[r * NUM_HEADS + h], p);
}

// ---------------------------------------------------------------------------
// values = mean over heads of p / s[row]; emit (row, col, values) flat.
// row/col are int32 written as raw bits into the output buffer.
// ---------------------------------------------------------------------------
__global__ void finalize_kernel(const int* __restrict__ ei,
                                const float* __restrict__ scores,
                                const float* __restrict__ sbuf,
                                int* __restrict__ out_row,
                                int* __restrict__ out_col,
                                float* __restrict__ out_val)
{
  int e = blockIdx.x * blockDim.x + threadIdx.x;
  if (e >= NUM_EDGES) return;
  int r = ei[e];
  int c = ei[NUM_EDGES + e];
  float p0 = scores[(size_t)e * 2 + 0];
  float p1 = scores[(size_t)e * 2 + 1];
  float v = 0.5f * (p0 / sbuf[r * 2 + 0] + p1 / sbuf[r * 2 + 1]);
  out_row[e] = r;
  out_col[e] = c;
  out_val[e] = v;
}

extern "C" void kernel_launch(void* const* d_in, const int* in_sizes, int n_in,
                              void* d_out, int out_size, void* d_ws, size_t ws_size,
                              hipStream_t stream) {
  const float* x  = (const float*)d_in[0];
  const int*   ei = (const int*)d_in[1];     // (2, NUM_EDGES) flat
  const float* W  = (const float*)d_in[2];
  const float* b  = (const float*)d_in[3];

  // Workspace layout (bytes): qk 51.2MB | scores 12.8MB | m 0.4MB | s 0.4MB
  char*  ws     = (char*)d_ws;
  float* qk     = (float*)(ws);
  float* scores = (float*)(ws + 51200000);
  float* mbuf   = (float*)(ws + 64000000);
  float* sbuf   = (float*)(ws + 64400000);

  int*   out_row = (int*)d_out;
  int*   out_col = out_row + NUM_EDGES;
  float* out_val = (float*)d_out + 2 * (size_t)NUM_EDGES;

  init_ms<<<(NUM_NODES * NUM_HEADS + 255) / 256, 256, 0, stream>>>(mbuf, sbuf);

  {
    const int waves = (NUM_NODES / 16) * (QK_COLS / 16);  // 50,000 tiles
    const int blocks = (waves + 7) / 8;                   // 8 waves / block
    gemm_qk_wmma<<<blocks, 256, 0, stream>>>(x, W, b, qk);
  }
  {
    const long long threads = (long long)NUM_EDGES * NUM_HEADS * 16;
    const int blocks = (int)((threads + 255) / 256);
    edge_scores_kernel<<<blocks, 256, 0, stream>>>(qk, ei, scores, mbuf);
  }
  exp_sum_kernel<<<(NUM_EDGES * NUM_HEADS + 255) / 256, 256, 0, stream>>>(
      ei, mbuf, scores, sbuf);
  finalize_kernel<<<(NUM_EDGES + 255) / 256, 256, 0, stream>>>(
      ei, scores, sbuf, out_row, out_col, out_val);
}